// DecoderIter_35837207118702
// MI455X (gfx1250) — compile-verified
//
#include <hip/hip_runtime.h>
#include <stdint.h>

// ---------------- problem dims ----------------
#define B_    256
#define T_    512
#define MEL_  80
#define PRE_  256
#define ENC_  512
#define ARNN_ 1024
#define ATT_  128
#define LF_   32
#define LK_   31
#define GQ_   4096   // 4*ARNN gate width

// ---------------- d_out layout (floats, return-order concat) ----------------
#define OFF_DEC    0
#define OFF_GATE   20480
#define OFF_AH     20736
#define OFF_AC     282880
#define OFF_DH     545024
#define OFF_DC     807168
#define OFF_AW     1069312
#define OFF_AWCUM  1200384
#define OFF_CTX    1331456

// ---------------- workspace layout (floats) ----------------
#define WS_PX1   0            // [256,256] prenet layer1 (dead after layer2)
#define WS_PX    65536        // [256,256] prenet out   (dead after arnn gates)
#define WS_G     131072       // [256,4096] gates (reused for arnn then drnn)
#define WS_PQ    1179648      // [256,128]
#define WS_LOC   1212416      // [256,512,32] (dead after energies)
#define WS_E     0            // [256,512] energies, reuses PX1+PX region
#define WS_CTXP  1212416      // [256,8,512] partial ctx, reuses LOC region

typedef __attribute__((ext_vector_type(2))) float v2f;
typedef __attribute__((ext_vector_type(8))) float v8f;

// ================= WMMA f32 16x16x4 (CDNA5) =================
__device__ __forceinline__ v8f wmma4(v2f a, v2f b, v8f c) {
  // D = A(16x4,f32) * B(4x16,f32) + C(16x16,f32)
  return __builtin_amdgcn_wmma_f32_16x16x4_f32(false, a, false, b, (short)0, c,
                                               false, false);
}

__device__ __forceinline__ v8f vzero8() {
  v8f z = {0.f, 0.f, 0.f, 0.f, 0.f, 0.f, 0.f, 0.f};
  return z;
}

// NT GEMM segment, single 16x16 tile.
__device__ __forceinline__ v8f gemm_seg(v8f acc, const float* __restrict__ a,
                                        const float* __restrict__ b, int K) {
#pragma unroll 4
  for (int k = 0; k < K; k += 4) {
    v2f av = *(const v2f*)(a + k);
    v2f bv = *(const v2f*)(b + k);
    acc = wmma4(av, bv, acc);
  }
  return acc;
}

// NT GEMM segment, 16x64 strip: reuse the A fragment across 4 N-tiles
// (5 loads per 4 wmma instead of 8).
__device__ __forceinline__ void gemm_seg4(v8f (&acc)[4],
                                          const float* __restrict__ a,
                                          const float* __restrict__ b, int lw,
                                          int K) {
  const float* b0 = b;
  const float* b1 = b + (size_t)16 * lw;
  const float* b2 = b + (size_t)32 * lw;
  const float* b3 = b + (size_t)48 * lw;
#pragma unroll 2
  for (int k = 0; k < K; k += 4) {
    v2f av = *(const v2f*)(a + k);
    acc[0] = wmma4(av, *(const v2f*)(b0 + k), acc[0]);
    acc[1] = wmma4(av, *(const v2f*)(b1 + k), acc[1]);
    acc[2] = wmma4(av, *(const v2f*)(b2 + k), acc[2]);
    acc[3] = wmma4(av, *(const v2f*)(b3 + k), acc[3]);
  }
}

// ================= threefry2x32 (JAX PRNG) =================
__device__ __forceinline__ void threefry2x32(uint32_t k0, uint32_t k1,
                                             uint32_t x0, uint32_t x1,
                                             uint32_t& o0, uint32_t& o1) {
  uint32_t ks0 = k0, ks1 = k1, ks2 = k0 ^ k1 ^ 0x1BD11BDAu;
  const int R[8] = {13, 15, 26, 6, 17, 29, 16, 24};
  x0 += ks0; x1 += ks1;
  uint32_t ks[3] = {ks0, ks1, ks2};
#pragma unroll
  for (int g = 0; g < 5; ++g) {
    const int* r = (g & 1) ? (R + 4) : R;
#pragma unroll
    for (int i = 0; i < 4; ++i) {
      x0 += x1;
      x1 = (x1 << r[i]) | (x1 >> (32 - r[i]));
      x1 ^= x0;
    }
    x0 += ks[(g + 1) % 3];
    x1 += ks[(g + 2) % 3] + (uint32_t)(g + 1);
  }
  o0 = x0; o1 = x1;
}

// prenet_infer mask: Bernoulli(0.5) per feature * 2.0, key = fold_in(key(42), li)
__device__ __forceinline__ float prenet_mask_scale(int li, unsigned j) {
  uint32_t fk0, fk1;
  threefry2x32(0u, 42u, 0u, (uint32_t)li, fk0, fk1);
  uint32_t i = (j < 128u) ? j : (j - 128u);
  uint32_t y0, y1;
  threefry2x32(fk0, fk1, i, i + 128u, y0, y1);
  uint32_t bits = (j < 128u) ? y0 : y1;
  float u = __uint_as_float((bits >> 9) | 0x3f800000u) - 1.0f;
  return (u <= 0.5f) ? 2.0f : 0.0f;
}

// ================= generic 16x16-tile WMMA GEMM =================
// Out[M,N] = X[M,K(ldx)] * W[N,K(ldw)]^T ; mode 1: relu * prenet_mask(li) * 2
__global__ void gemm16_kernel(const float* __restrict__ X, int ldx,
                              const float* __restrict__ W, int ldw, int Ntiles,
                              int K, float* __restrict__ Out, int ldo, int mode,
                              int li) {
  unsigned lane = threadIdx.x & 31u;
  unsigned widx = threadIdx.x >> 5;
  unsigned gw = blockIdx.x * 8u + widx;
  unsigned bt = gw / (unsigned)Ntiles;
  unsigned nt = gw % (unsigned)Ntiles;
  unsigned row = lane & 15u;
  unsigned kk = (lane >> 4) * 2u;

  const float* a = X + (size_t)(bt * 16u + row) * ldx + kk;
  const float* b = W + (size_t)(nt * 16u + row) * ldw + kk;
  v8f acc = vzero8();
  acc = gemm_seg(acc, a, b, K);

  unsigned n = nt * 16u + row;
  unsigned mrow = (lane >> 4) * 8u;
  float scale = (mode == 1) ? prenet_mask_scale(li, n) : 1.0f;
#pragma unroll
  for (int r = 0; r < 8; ++r) {
    float v = acc[r];
    if (mode == 1) v = fmaxf(v, 0.0f) * scale;
    Out[(size_t)(bt * 16u + mrow + r) * ldo + n] = v;
  }
}

// ================= fused 3-segment LSTM gate GEMM =================
// G[B,4096] = X0*W0^T + X1*W1^T + X2*W2^T + b0 + b1
// One wave computes a 16(m) x 64(o) strip (4 accumulators, A-frag reuse).
__global__ void lstm_gates_kernel(const float* __restrict__ X0, int ld0,
                                  const float* __restrict__ W0, int lw0, int K0,
                                  const float* __restrict__ X1, int ld1,
                                  const float* __restrict__ W1, int lw1, int K1,
                                  const float* __restrict__ X2, int ld2,
                                  const float* __restrict__ W2, int lw2, int K2,
                                  const float* __restrict__ b0,
                                  const float* __restrict__ b1,
                                  float* __restrict__ G) {
  unsigned lane = threadIdx.x & 31u;
  unsigned widx = threadIdx.x >> 5;
  unsigned o0 = (blockIdx.x * 8u + widx) * 64u;  // grid.x=8 -> o in [0,4096)
  unsigned m0 = blockIdx.y * 16u;                // 0..240
  unsigned row = lane & 15u;
  unsigned kk = (lane >> 4) * 2u;

  v8f acc[4];
#pragma unroll
  for (int j = 0; j < 4; ++j) acc[j] = vzero8();

  gemm_seg4(acc, X0 + (size_t)(m0 + row) * ld0 + kk,
            W0 + (size_t)(o0 + row) * lw0 + kk, lw0, K0);
  gemm_seg4(acc, X1 + (size_t)(m0 + row) * ld1 + kk,
            W1 + (size_t)(o0 + row) * lw1 + kk, lw1, K1);
  gemm_seg4(acc, X2 + (size_t)(m0 + row) * ld2 + kk,
            W2 + (size_t)(o0 + row) * lw2 + kk, lw2, K2);

  unsigned mrow = (lane >> 4) * 8u;
#pragma unroll
  for (int j = 0; j < 4; ++j) {
    unsigned n = o0 + j * 16u + row;
    float bias = b0[n] + b1[n];
#pragma unroll
    for (int r = 0; r < 8; ++r)
      G[(size_t)(m0 + mrow + r) * GQ_ + n] = acc[j][r] + bias;
  }
}

// ================= LSTM pointwise (gate order i,f,g,o) =================
__global__ void lstm_point_kernel(const float* __restrict__ G,
                                  const float* __restrict__ cprev,
                                  float* __restrict__ h, float* __restrict__ c) {
  size_t tid = (size_t)blockIdx.x * blockDim.x + threadIdx.x;  // B*1024
  unsigned b = (unsigned)(tid >> 10);
  unsigned j = (unsigned)(tid & 1023u);
  const float* g = G + (size_t)b * GQ_;
  float gi = 1.0f / (1.0f + expf(-g[j]));
  float gf = 1.0f / (1.0f + expf(-g[1024 + j]));
  float gg = tanhf(g[2048 + j]);
  float go = 1.0f / (1.0f + expf(-g[3072 + j]));
  float c2 = gf * cprev[tid] + gi * gg;
  c[tid] = c2;
  h[tid] = go * tanhf(c2);
}

// ================= location conv: loc[b,t,f] =================
__global__ void loc_conv_kernel(const float* __restrict__ aw,
                                const float* __restrict__ awc,
                                const float* __restrict__ w,
                                float* __restrict__ loc) {
  int f = threadIdx.x & 31;
  int t = blockIdx.x * 8 + (threadIdx.x >> 5);
  int b = blockIdx.y;
  const float* a0 = aw + (size_t)b * T_;
  const float* a1 = awc + (size_t)b * T_;
  const float* wf = w + (size_t)f * (2 * LK_);  // [f][0][k], [f][1][k]
  float acc = 0.f;
#pragma unroll
  for (int k = 0; k < LK_; ++k) {
    int ts = t + k - (LK_ - 1) / 2;
    if (ts >= 0 && ts < T_) acc += a0[ts] * wf[k] + a1[ts] * wf[LK_ + k];
  }
  loc[((size_t)b * T_ + t) * LF_ + f] = acc;
}

// ================= energies: WMMA ploc + tanh + dot(v) =================
// one wave -> 16 t-values of one batch row
__global__ void energies_kernel(const float* __restrict__ loc,
                                const float* __restrict__ llw,  // [ATT,LF]
                                const float* __restrict__ pq,   // [B,ATT]
                                const float* __restrict__ pm,   // [B,T,ATT]
                                const float* __restrict__ v,    // [ATT]
                                float* __restrict__ e) {
  unsigned lane = threadIdx.x & 31u;
  unsigned widx = threadIdx.x >> 5;
  unsigned gw = blockIdx.x * 8u + widx;  // 0..8191
  unsigned b = gw >> 5;
  unsigned t0 = (gw & 31u) * 16u;
  unsigned row = lane & 15u;
  unsigned half = lane >> 4;
  unsigned kk = half * 2u;

  const float* aPtr = loc + ((size_t)b * T_ + t0 + row) * LF_ + kk;
  v8f acc[8];
#pragma unroll
  for (int nt = 0; nt < 8; ++nt) acc[nt] = vzero8();

#pragma unroll
  for (int k = 0; k < LF_; k += 4) {
    v2f a = *(const v2f*)(aPtr + k);
#pragma unroll
    for (int nt = 0; nt < 8; ++nt) {
      v2f bb = *(const v2f*)(llw + (size_t)(nt * 16 + row) * LF_ + kk + k);
      acc[nt] = wmma4(a, bb, acc[nt]);
    }
  }

  float part[8] = {0.f, 0.f, 0.f, 0.f, 0.f, 0.f, 0.f, 0.f};
#pragma unroll
  for (int nt = 0; nt < 8; ++nt) {
    unsigned aIdx = nt * 16u + row;
    float pqv = pq[(size_t)b * ATT_ + aIdx];
    float vv = v[aIdx];
#pragma unroll
    for (int r = 0; r < 8; ++r) {
      unsigned t = t0 + r + half * 8u;
      float pmv = pm[((size_t)b * T_ + t) * ATT_ + aIdx];
      part[r] += vv * tanhf(acc[nt][r] + pqv + pmv);
    }
  }
  // reduce over 16 lanes in each half (a-dimension)
#pragma unroll
  for (int m = 8; m; m >>= 1)
#pragma unroll
    for (int r = 0; r < 8; ++r) part[r] += __shfl_xor(part[r], m, 16);

  if (row == 0) {
#pragma unroll
    for (int r = 0; r < 8; ++r)
      e[(size_t)b * T_ + t0 + r + half * 8u] = part[r];
  }
}

// ================= masked softmax + cum =================
__global__ void softmax_kernel(const float* __restrict__ e,
                               const unsigned char* __restrict__ mask,
                               const float* __restrict__ awcum_in,
                               float* __restrict__ aw_out,
                               float* __restrict__ awcum_out) {
  __shared__ float red[16];
  __shared__ float bsh[2];
  int b = blockIdx.x;
  int t = threadIdx.x;  // 512 threads
  size_t idx = (size_t)b * T_ + t;
  float x = e[idx];
  if (mask[idx]) x = -__builtin_inff();

  float m = x;
#pragma unroll
  for (int off = 16; off; off >>= 1) m = fmaxf(m, __shfl_xor(m, off, 32));
  if ((t & 31) == 0) red[t >> 5] = m;
  __syncthreads();
  if (t < 32) {
    float vv = (t < 16) ? red[t] : -__builtin_inff();
#pragma unroll
    for (int off = 8; off; off >>= 1) vv = fmaxf(vv, __shfl_xor(vv, off, 32));
    if (t == 0) bsh[0] = vv;
  }
  __syncthreads();
  float p = expf(x - bsh[0]);
  if (!(x > -__builtin_inff())) p = 0.f;

  float s = p;
#pragma unroll
  for (int off = 16; off; off >>= 1) s += __shfl_xor(s, off, 32);
  if ((t & 31) == 0) red[t >> 5] = s;
  __syncthreads();
  if (t < 32) {
    float vv = (t < 16) ? red[t] : 0.f;
#pragma unroll
    for (int off = 8; off; off >>= 1) vv += __shfl_xor(vv, off, 32);
    if (t == 0) bsh[1] = vv;
  }
  __syncthreads();
  float aw = p / bsh[1];
  aw_out[idx] = aw;
  awcum_out[idx] = awcum_in[idx] + aw;
}

// ================= ctx = aw @ memory (streams 268MB -> HBM bound) ==========
__global__ void ctx_part_kernel(const float* __restrict__ aw,
                                const float* __restrict__ mem,
                                float* __restrict__ ctxp) {
  int b = blockIdx.x, ch = blockIdx.y, tid = threadIdx.x;  // 128 threads
  const float* base = mem + (size_t)b * T_ * ENC_;
  float4 acc = make_float4(0.f, 0.f, 0.f, 0.f);
  int t0 = ch * 64;
  for (int t = t0; t < t0 + 64; ++t) {
    float w = aw[(size_t)b * T_ + t];
    if (t + 4 < t0 + 64)
      __builtin_prefetch(base + (size_t)(t + 4) * ENC_ + tid * 4, 0, 0);
    float4 m4 = *(const float4*)(base + (size_t)t * ENC_ + tid * 4);
    acc.x += w * m4.x; acc.y += w * m4.y;
    acc.z += w * m4.z; acc.w += w * m4.w;
  }
  *((float4*)(ctxp + ((size_t)b * 8 + ch) * ENC_) + tid) = acc;
}

__global__ void ctx_red_kernel(const float* __restrict__ ctxp,
                               float* __restrict__ ctx) {
  int tid = blockIdx.x * blockDim.x + threadIdx.x;  // B*ENC
  int b = tid >> 9, ee = tid & 511;
  float s = 0.f;
#pragma unroll
  for (int c = 0; c < 8; ++c) s += ctxp[((size_t)b * 8 + c) * ENC_ + ee];
  ctx[(size_t)b * ENC_ + ee] = s;
}

// ================= projection + gate =================
__global__ void projgate_kernel(const float* __restrict__ dh,
                                const float* __restrict__ ctx,
                                const float* __restrict__ pw,
                                const float* __restrict__ pb,
                                const float* __restrict__ gw,
                                const float* __restrict__ gb,
                                float* __restrict__ dec,
                                float* __restrict__ gate) {
  int b = blockIdx.x;
  int o = threadIdx.x;  // 96 threads, 81 active
  if (o >= 81) return;
  const float* w = (o < 80) ? (pw + (size_t)o * 1536) : gw;
  const float* h = dh + (size_t)b * 1024;
  const float* c = ctx + (size_t)b * 512;
  float acc = 0.f;
  for (int k = 0; k < 1024; ++k) acc += h[k] * w[k];
  for (int k = 0; k < 512; ++k) acc += c[k] * w[1024 + k];
  if (o < 80) dec[(size_t)b * MEL_ + o] = acc + pb[o];
  else        gate[b] = acc + gb[0];
}

// ================= host orchestration =================
extern "C" void kernel_launch(void* const* d_in, const int* in_sizes, int n_in,
                              void* d_out, int out_size, void* d_ws,
                              size_t ws_size, hipStream_t stream) {
  (void)in_sizes; (void)n_in; (void)out_size; (void)ws_size;
  const float* dec_in   = (const float*)d_in[0];
  const float* ah_in    = (const float*)d_in[1];
  const float* ac_in    = (const float*)d_in[2];
  const float* dh_in    = (const float*)d_in[3];
  const float* dc_in    = (const float*)d_in[4];
  const float* aw_in    = (const float*)d_in[5];
  const float* awcum_in = (const float*)d_in[6];
  const float* attctx   = (const float*)d_in[7];
  const float* memory   = (const float*)d_in[8];
  const float* pm       = (const float*)d_in[9];
  const unsigned char* mask = (const unsigned char*)d_in[10];
  const float* pw1      = (const float*)d_in[11];
  const float* pw2      = (const float*)d_in[12];
  const float* a_wih    = (const float*)d_in[13];
  const float* a_whh    = (const float*)d_in[14];
  const float* a_bih    = (const float*)d_in[15];
  const float* a_bhh    = (const float*)d_in[16];
  const float* q_w      = (const float*)d_in[17];
  const float* lconv_w  = (const float*)d_in[18];
  const float* llin_w   = (const float*)d_in[19];
  const float* v_w      = (const float*)d_in[20];
  const float* d_wih    = (const float*)d_in[21];
  const float* d_whh    = (const float*)d_in[22];
  const float* d_bih    = (const float*)d_in[23];
  const float* d_bhh    = (const float*)d_in[24];
  const float* proj_w   = (const float*)d_in[25];
  const float* proj_b   = (const float*)d_in[26];
  const float* gate_w   = (const float*)d_in[27];
  const float* gate_b   = (const float*)d_in[28];

  float* W = (float*)d_ws;
  float* px1  = W + WS_PX1;
  float* px   = W + WS_PX;
  float* G    = W + WS_G;
  float* pq   = W + WS_PQ;
  float* loc  = W + WS_LOC;
  float* e    = W + WS_E;
  float* ctxp = W + WS_CTXP;

  float* out    = (float*)d_out;
  float* o_dec  = out + OFF_DEC;
  float* o_gate = out + OFF_GATE;
  float* o_ah   = out + OFF_AH;
  float* o_ac   = out + OFF_AC;
  float* o_dh   = out + OFF_DH;
  float* o_dc   = out + OFF_DC;
  float* o_aw   = out + OFF_AW;
  float* o_awc  = out + OFF_AWCUM;
  float* o_ctx  = out + OFF_CTX;

  // 1-2. prenet (relu + fixed Bernoulli mask * 2)
  gemm16_kernel<<<32, 256, 0, stream>>>(dec_in, MEL_, pw1, MEL_, 16, MEL_, px1,
                                        PRE_, 1, 0);
  gemm16_kernel<<<32, 256, 0, stream>>>(px1, PRE_, pw2, PRE_, 16, PRE_, px,
                                        PRE_, 1, 1);
  // 3. attention-RNN gates: [px | attctx] @ w_ih^T + ah @ w_hh^T + biases
  lstm_gates_kernel<<<dim3(8, 16), 256, 0, stream>>>(
      px, PRE_, a_wih, PRE_ + ENC_, PRE_,
      attctx, ENC_, a_wih + PRE_, PRE_ + ENC_, ENC_,
      ah_in, ARNN_, a_whh, ARNN_, ARNN_,
      a_bih, a_bhh, G);
  // 4. attention-RNN pointwise
  lstm_point_kernel<<<1024, 256, 0, stream>>>(G, ac_in, o_ah, o_ac);
  // 5. pq = ah @ q_w^T
  gemm16_kernel<<<16, 256, 0, stream>>>(o_ah, ARNN_, q_w, ARNN_, 8, ARNN_, pq,
                                        ATT_, 0, 0);
  // 6. location conv
  loc_conv_kernel<<<dim3(64, 256), 256, 0, stream>>>(aw_in, awcum_in, lconv_w,
                                                     loc);
  // 7. energies = tanh(pq + ploc + pm) . v  (WMMA over f->ATT)
  energies_kernel<<<1024, 256, 0, stream>>>(loc, llin_w, pq, pm, v_w, e);
  // 8. masked softmax + cumulative weights
  softmax_kernel<<<256, 512, 0, stream>>>(e, mask, awcum_in, o_aw, o_awc);
  // 9-10. context vector
  ctx_part_kernel<<<dim3(256, 8), 128, 0, stream>>>(o_aw, memory, ctxp);
  ctx_red_kernel<<<512, 256, 0, stream>>>(ctxp, o_ctx);
  // 11. decoder-RNN gates: [ah | ctx] @ w_ih^T + dh @ w_hh^T + biases
  lstm_gates_kernel<<<dim3(8, 16), 256, 0, stream>>>(
      o_ah, ARNN_, d_wih, ARNN_ + ENC_, ARNN_,
      o_ctx, ENC_, d_wih + ARNN_, ARNN_ + ENC_, ENC_,
      dh_in, ARNN_, d_whh, ARNN_, ARNN_,
      d_bih, d_bhh, G);
  // 12. decoder-RNN pointwise
  lstm_point_kernel<<<1024, 256, 0, stream>>>(G, dc_in, o_dh, o_dc);
  // 13. projection + gate
  projgate_kernel<<<256, 96, 0, stream>>>(o_dh, o_ctx, proj_w, proj_b, gate_w,
                                          gate_b, o_dec, o_gate);
}